// Encoder_111669149946
// MI455X (gfx1250) — compile-verified
//
#include <hip/hip_runtime.h>
#include <hip/hip_bf16.h>

typedef __attribute__((ext_vector_type(2))) float v2f;
typedef __attribute__((ext_vector_type(8))) float v8f;

#define IN_CH  128
#define HID_CH 64
#define OUT_CH 32

// ---------------------------------------------------------------------------
// Elementwise / setup kernels
// ---------------------------------------------------------------------------

__global__ void k_zero(float* __restrict__ p, int n) {
    int i = blockIdx.x * blockDim.x + threadIdx.x;
    if (i < n) p[i] = 0.0f;
}

__global__ void k_degree(const int* __restrict__ dst, float* __restrict__ deg, int nE) {
    int e = blockIdx.x * blockDim.x + threadIdx.x;
    if (e < nE) atomicAdd(&deg[dst[e]], 1.0f);
}

__global__ void k_rsqrt(float* __restrict__ dis, int n) {
    int i = blockIdx.x * blockDim.x + threadIdx.x;
    if (i < n) dis[i] = rsqrtf(dis[i] + 1.0f);   // +1 self loop, deg >= 1
}

__global__ void k_edgew(const int* __restrict__ src, const int* __restrict__ dst,
                        const float* __restrict__ dis, float* __restrict__ ew, int nE) {
    int e = blockIdx.x * blockDim.x + threadIdx.x;
    if (e < nE) ew[e] = dis[src[e]] * dis[dst[e]];
}

// W3[k, 0:32] = W_mu[k, :], W3[k, 32:64] = W_ls[k, :]
__global__ void k_packw3(const float* __restrict__ Wmu, const float* __restrict__ Wls,
                         float* __restrict__ W3) {
    int t = blockIdx.x * blockDim.x + threadIdx.x;  // 64*32 threads
    if (t >= HID_CH * OUT_CH) return;
    int k = t >> 5, c = t & 31;
    W3[k * 64 + c]      = Wmu[k * 32 + c];
    W3[k * 64 + 32 + c] = Wls[k * 32 + c];
}

// ---------------------------------------------------------------------------
// GEMM: C[N,64] = A[N,K] @ B[K,64] using V_WMMA_F32_16X16X4_F32.
// One wave -> one 16x16 tile of C. Block = 128 threads = 4 waves covering the
// 64 output columns; grid.x covers row tiles.
//
// A-frag (16x4 f32, ISA 7.12.2): lane l holds rows m=l&15, k-pair (l>>4)*2.
// B-frag (4x16): symmetric layout assumed: lane l holds col n=l&15, same k-pair.
// C/D (16x16 f32): lane l col n=l&15, VGPR v -> row v + 8*(l>>4).
// ---------------------------------------------------------------------------
template <int K>
__global__ __launch_bounds__(128) void k_gemm_wmma(const float* __restrict__ A,
                                                   const float* __restrict__ B,
                                                   float* __restrict__ C, int N) {
    const int lane = threadIdx.x & 31;
    const int wave = threadIdx.x >> 5;
    const int r0 = blockIdx.x * 16;
    const int c0 = wave * 16;
    const int ml = lane & 15;
    const int kh = (lane >> 4) * 2;     // 0 or 2

    int ra = r0 + ml;
    if (ra >= N) ra = N - 1;            // clamp (N is a multiple of 16 anyway)
    const float* Arow = A + (long)ra * K;
    const float* Bcol = B + c0 + ml;

    v8f acc = {};
    #pragma unroll
    for (int k = 0; k < K; k += 4) {
        v2f a = *(const v2f*)(Arow + k + kh);           // b64 load, 8B aligned
        v2f b;
        b.x = Bcol[(k + kh) * 64];
        b.y = Bcol[(k + kh + 1) * 64];
        acc = __builtin_amdgcn_wmma_f32_16x16x4_f32(
            /*neg_a=*/false, a, /*neg_b=*/false, b,
            /*c_mod=*/(short)0, acc, /*reuse_a=*/false, /*reuse_b=*/false);
    }

    const int n = c0 + ml;
    const int mbase = r0 + 8 * (lane >> 4);
    #pragma unroll
    for (int v = 0; v < 8; ++v) {
        int m = mbase + v;
        if (m < N) C[(long)m * 64 + n] = acc[v];
    }
}

// ---------------------------------------------------------------------------
// Aggregation kernels (C = 64 channels everywhere after the GEMM)
// ---------------------------------------------------------------------------

// agg[i,c] = t[i,c] * dis[i]^2   (self-loop term; also serves as the init)
__global__ void k_selfinit(const float* __restrict__ t, const float* __restrict__ dis,
                           float* __restrict__ agg, int N) {
    int tid = blockIdx.x * blockDim.x + threadIdx.x;   // N*64 threads
    if (tid >= N * 64) return;
    int i = tid >> 6;
    float d = dis[i];
    agg[tid] = t[tid] * d * d;
}

// For each edge e: agg[dst[e], :] += t[src[e], :] * ew[e]
// 16 threads per edge, float4 gather (b128), 4 f32 atomics each.
__global__ void k_scatter(const float* __restrict__ t, const int* __restrict__ src,
                          const int* __restrict__ dst, const float* __restrict__ ew,
                          float* __restrict__ agg, int nE) {
    int tid = blockIdx.x * blockDim.x + threadIdx.x;
    int e = tid >> 4;
    if (e >= nE) return;
    int sub = tid & 15;
    int s = src[e];
    int d = dst[e];
    float w = ew[e];
    float4 v = *(const float4*)(t + (long)s * 64 + sub * 4);
    float* ap = agg + (long)d * 64 + sub * 4;
    atomicAdd(ap + 0, v.x * w);
    atomicAdd(ap + 1, v.y * w);
    atomicAdd(ap + 2, v.z * w);
    atomicAdd(ap + 3, v.w * w);
}

// h[i,c] = leaky_relu(agg[i,c] + b[c], 0.01)
__global__ void k_bias_leaky(const float* __restrict__ agg, const float* __restrict__ b,
                             float* __restrict__ h, int N) {
    int tid = blockIdx.x * blockDim.x + threadIdx.x;
    if (tid >= N * 64) return;
    float v = agg[tid] + b[tid & 63];
    h[tid] = fmaxf(v, 0.01f * v);
}

// Split concatenated agg[N,64] into mu = agg[:, :32]+b_mu, ls = agg[:, 32:]+b_ls
__global__ void k_final(const float* __restrict__ agg, const float* __restrict__ bmu,
                        const float* __restrict__ bls, float* __restrict__ out, int N) {
    int tid = blockIdx.x * blockDim.x + threadIdx.x;   // N*32 threads
    if (tid >= N * 32) return;
    int i = tid >> 5, c = tid & 31;
    out[tid]          = agg[(long)i * 64 + c]      + bmu[c];
    out[N * 32 + tid] = agg[(long)i * 64 + 32 + c] + bls[c];
}

// ---------------------------------------------------------------------------
// Host driver
// ---------------------------------------------------------------------------
static inline int cdiv(long a, int b) { return (int)((a + b - 1) / b); }

extern "C" void kernel_launch(void* const* d_in, const int* in_sizes, int n_in,
                              void* d_out, int out_size, void* d_ws, size_t ws_size,
                              hipStream_t stream) {
    const float* x   = (const float*)d_in[0];
    const int*   ei  = (const int*)d_in[1];
    const float* W1  = (const float*)d_in[2];
    const float* b1  = (const float*)d_in[3];
    const float* W2  = (const float*)d_in[4];
    const float* b2  = (const float*)d_in[5];
    const float* Wmu = (const float*)d_in[6];
    const float* bmu = (const float*)d_in[7];
    const float* Wls = (const float*)d_in[8];
    const float* bls = (const float*)d_in[9];

    const int N = in_sizes[0] / IN_CH;   // 100000
    const int E = in_sizes[1] / 2;       // 1600000
    const int* src = ei;
    const int* dst = ei + E;

    // Scratch layout (floats): dis[N] | ew[E] | W3[4096] | buf1[N*64] | buf2[N*64]
    float* p    = (float*)d_ws;
    float* dis  = p;               p += N;
    float* ew   = p;               p += E;
    float* W3   = p;               p += 64 * 64;
    float* buf1 = p;               p += (size_t)N * 64;
    float* buf2 = p;

    const int T = 256;
    const int gN   = cdiv(N, T);
    const int gE   = cdiv(E, T);
    const int gNC  = cdiv((long)N * 64, T);
    const int gE16 = cdiv((long)E * 16, T);
    const int gOut = cdiv((long)N * 32, T);
    const int gGemm = cdiv(N, 16);

    // --- normalization ---
    k_zero  <<<gN, T, 0, stream>>>(dis, N);
    k_degree<<<gE, T, 0, stream>>>(dst, dis, E);
    k_rsqrt <<<gN, T, 0, stream>>>(dis, N);
    k_edgew <<<gE, T, 0, stream>>>(src, dst, dis, ew, E);
    k_packw3<<<cdiv(HID_CH * OUT_CH, T), T, 0, stream>>>(Wmu, Wls, W3);

    // --- conv1: x[N,128] @ W1 -> buf1; aggregate -> buf2; h1 -> buf1 ---
    k_gemm_wmma<IN_CH><<<gGemm, 128, 0, stream>>>(x, W1, buf1, N);
    k_selfinit  <<<gNC,  T, 0, stream>>>(buf1, dis, buf2, N);
    k_scatter   <<<gE16, T, 0, stream>>>(buf1, src, dst, ew, buf2, E);
    k_bias_leaky<<<gNC,  T, 0, stream>>>(buf2, b1, buf1, N);

    // --- conv2: buf1[N,64] @ W2 -> buf2; aggregate -> buf1; h2 -> buf2 ---
    k_gemm_wmma<HID_CH><<<gGemm, 128, 0, stream>>>(buf1, W2, buf2, N);
    k_selfinit  <<<gNC,  T, 0, stream>>>(buf2, dis, buf1, N);
    k_scatter   <<<gE16, T, 0, stream>>>(buf2, src, dst, ew, buf1, E);
    k_bias_leaky<<<gNC,  T, 0, stream>>>(buf1, b2, buf2, N);

    // --- heads fused: buf2[N,64] @ [Wmu|Wls] -> buf1; aggregate -> buf2 ---
    k_gemm_wmma<HID_CH><<<gGemm, 128, 0, stream>>>(buf2, W3, buf1, N);
    k_selfinit<<<gNC,  T, 0, stream>>>(buf1, dis, buf2, N);
    k_scatter <<<gE16, T, 0, stream>>>(buf1, src, dst, ew, buf2, E);
    k_final   <<<gOut, T, 0, stream>>>(buf2, bmu, bls, (float*)d_out, N);
}